// DCGRU_37898791420595
// MI455X (gfx1250) — compile-verified
//
#include <hip/hip_runtime.h>
#include <hip/hip_bf16.h>
#include <math.h>

#define TT 8
#define NN 20000
#define EE 640000
#define DIN 2
#define HH 64
#define FIN 66
#define FPAD 68
#define MTILES (NN/16)

typedef __attribute__((ext_vector_type(2))) float v2f;
typedef __attribute__((ext_vector_type(8))) float v8f;

// ---------------- elementwise / graph-norm kernels ----------------

__global__ void fill_kernel(float* __restrict__ p, int n, float v) {
  int i = blockIdx.x * blockDim.x + threadIdx.x;
  if (i < n) p[i] = v;
}

__global__ void edge_deg_kernel(const int* __restrict__ src, const float* __restrict__ w,
                                float* __restrict__ deg) {
  int e = blockIdx.x * blockDim.x + threadIdx.x;
  if (e < EE) atomicAdd(&deg[src[e]], w[e]);
}

__global__ void dinv_kernel(float* __restrict__ deg) {
  int i = blockIdx.x * blockDim.x + threadIdx.x;
  if (i < NN) { float d = deg[i]; deg[i] = (d > 0.f) ? rsqrtf(d) : 0.f; }
}

__global__ void wn_kernel(const int* __restrict__ src, const int* __restrict__ dst,
                          const float* __restrict__ w, const float* __restrict__ dinv,
                          float* __restrict__ wn) {
  int e = blockIdx.x * blockDim.x + threadIdx.x;
  if (e < EE) wn[e] = -dinv[src[e]] * w[e] * dinv[dst[e]];
}

// comb = [x_t, h] (useR=0) or [x_t, r*h] (useR=1), padded with zeros to FPAD cols
__global__ void build_comb_kernel(const float* __restrict__ xt, const float* __restrict__ h,
                                  const float* __restrict__ rg, int useR,
                                  float* __restrict__ C) {
  int i = blockIdx.x * blockDim.x + threadIdx.x;
  if (i >= NN * FPAD) return;
  int n = i / FPAD, f = i - n * FPAD;
  float v = 0.f;
  if (f < DIN) {
    v = xt[n * DIN + f];
  } else if (f < FIN) {
    float hv = h[n * HH + (f - DIN)];
    v = useR ? rg[n * HH + (f - DIN)] * hv : hv;
  }
  C[i] = v;
}

// out[dst] += wn[e] * in[src] over 66 features; one wave32 per edge
__global__ void spmm_kernel(const int* __restrict__ src, const int* __restrict__ dst,
                            const float* __restrict__ wn, const float* __restrict__ in,
                            float* __restrict__ out) {
  int e = blockIdx.x * (blockDim.x >> 5) + (threadIdx.x >> 5);
  if (e >= EE) return;
  int lane = threadIdx.x & 31;
  int s = src[e], d = dst[e];
  float w = wn[e];
  const float* pi = in + (size_t)s * FPAD;
  float* po = out + (size_t)d * FPAD;
  atomicAdd(&po[lane],        w * pi[lane]);
  atomicAdd(&po[lane + 32],   w * pi[lane + 32]);   // 32..63 < 66
  if (lane < (FIN - 64)) atomicAdd(&po[lane + 64], w * pi[lane + 64]); // 64,65
}

// Tx2 = 2*prop(Tx1) - Tx0   (pad cols stay 0: 2*0-0)
__global__ void finish_t2_kernel(float* __restrict__ T2, const float* __restrict__ C) {
  int i = blockIdx.x * blockDim.x + threadIdx.x;
  if (i < NN * FPAD) T2[i] = 2.f * T2[i] - C[i];
}

// ---------------- WMMA fp32 GEMM kernels ----------------
// One wave computes a 16x64 output tile: D = sum_p Tx_p @ W[p], K padded 66->68.

__device__ inline void gemm_hop(const float* __restrict__ A, const float* __restrict__ W,
                                int row0, int ma, int kh, v8f acc[4]) {
  const float* ap = A + (size_t)(row0 + ma) * FPAD + 2 * kh;
  for (int k0 = 0; k0 < FPAD; k0 += 4) {
    v2f a;
    a.x = ap[k0];
    a.y = ap[k0 + 1];
    int kb = k0 + 2 * kh;
    bool ok0 = kb < FIN, ok1 = (kb + 1) < FIN;
#pragma unroll
    for (int nb = 0; nb < 4; ++nb) {
      int col = nb * 16 + ma;
      v2f b;
      b.x = ok0 ? W[kb * HH + col] : 0.f;
      b.y = ok1 ? W[(kb + 1) * HH + col] : 0.f;
      acc[nb] = __builtin_amdgcn_wmma_f32_16x16x4_f32(
          false, a, false, b, (short)0, acc[nb], false, false);
    }
  }
}

__launch_bounds__(256)
__global__ void gemm_ru_kernel(const float* __restrict__ A0, const float* __restrict__ A1,
                               const float* __restrict__ A2,
                               const float* __restrict__ Wr, const float* __restrict__ br,
                               const float* __restrict__ Wu, const float* __restrict__ bu,
                               float* __restrict__ rO, float* __restrict__ uO) {
  int wave = blockIdx.x * 8 + (threadIdx.x >> 5);
  if (wave >= MTILES) return;            // wave-uniform: EXEC all-ones inside
  int lane = threadIdx.x & 31;
  int ma = lane & 15;
  int kh = lane >> 4;
  int row0 = wave * 16;
  v8f accR[4] = {}; v8f accU[4] = {};
  const float* As[3] = {A0, A1, A2};
#pragma unroll
  for (int p = 0; p < 3; ++p) {
    const float* Ap = (p == 0) ? A0 : ((p == 1) ? A1 : A2);
    (void)As;
    gemm_hop(Ap, Wr + p * FIN * HH, row0, ma, kh, accR);
    gemm_hop(Ap, Wu + p * FIN * HH, row0, ma, kh, accU);
  }
#pragma unroll
  for (int nb = 0; nb < 4; ++nb) {
    int col = nb * 16 + ma;
    float biasR = br[col], biasU = bu[col];
#pragma unroll
    for (int ri = 0; ri < 8; ++ri) {
      int row = row0 + ri + kh * 8;
      float zr = accR[nb][ri] + biasR;
      float zu = accU[nb][ri] + biasU;
      rO[row * HH + col] = 1.f / (1.f + __expf(-zr));
      uO[row * HH + col] = 1.f / (1.f + __expf(-zu));
    }
  }
}

__launch_bounds__(256)
__global__ void gemm_c_kernel(const float* __restrict__ A0, const float* __restrict__ A1,
                              const float* __restrict__ A2,
                              const float* __restrict__ Wc, const float* __restrict__ bc,
                              const float* __restrict__ uG, float* __restrict__ h) {
  int wave = blockIdx.x * 8 + (threadIdx.x >> 5);
  if (wave >= MTILES) return;
  int lane = threadIdx.x & 31;
  int ma = lane & 15;
  int kh = lane >> 4;
  int row0 = wave * 16;
  v8f acc[4] = {};
#pragma unroll
  for (int p = 0; p < 3; ++p) {
    const float* Ap = (p == 0) ? A0 : ((p == 1) ? A1 : A2);
    gemm_hop(Ap, Wc + p * FIN * HH, row0, ma, kh, acc);
  }
#pragma unroll
  for (int nb = 0; nb < 4; ++nb) {
    int col = nb * 16 + ma;
    float bias = bc[col];
#pragma unroll
    for (int ri = 0; ri < 8; ++ri) {
      int row = row0 + ri + kh * 8;
      float c = tanhf(acc[nb][ri] + bias);
      float uu = uG[row * HH + col];
      float ho = h[row * HH + col];
      h[row * HH + col] = uu * ho + (1.f - uu) * c;   // in-place, 1 reader/writer per elem
    }
  }
}

// ---------------- host-side sequencing ----------------

extern "C" void kernel_launch(void* const* d_in, const int* in_sizes, int n_in,
                              void* d_out, int out_size, void* d_ws, size_t ws_size,
                              hipStream_t stream) {
  const float* x     = (const float*)d_in[0];
  const int*   eidx  = (const int*)d_in[1];
  const float* eattr = (const float*)d_in[2];
  const float* Wr    = (const float*)d_in[3];
  const float* br    = (const float*)d_in[4];
  const float* Wu    = (const float*)d_in[5];
  const float* bu    = (const float*)d_in[6];
  const float* Wc    = (const float*)d_in[7];
  const float* bc    = (const float*)d_in[8];
  float* h  = (float*)d_out;
  float* ws = (float*)d_ws;

  size_t off = 0;
  float* wn  = ws + off; off += (size_t)EE;
  float* deg = ws + off; off += (size_t)NN;
  float* C   = ws + off; off += (size_t)NN * FPAD;
  float* T1  = ws + off; off += (size_t)NN * FPAD;
  float* T2  = ws + off; off += (size_t)NN * FPAD;
  float* rG  = ws + off; off += (size_t)NN * HH;
  float* uG  = ws + off; off += (size_t)NN * HH;

  const int B = 256;
  int gN  = (NN + B - 1) / B;
  int gE  = (EE + B - 1) / B;
  int gNF = (NN * FPAD + B - 1) / B;
  int gNH = (NN * HH + B - 1) / B;
  int gSp = (EE + 7) / 8;          // 8 waves per block, one edge per wave
  int gGm = (MTILES + 7) / 8;

  fill_kernel<<<gNH, B, 0, stream>>>(h, NN * HH, 0.f);   // h0 = 0

  for (int t = 0; t < TT; ++t) {
    const int*   src = eidx + (size_t)t * 2 * EE;
    const int*   dst = src + EE;
    const float* w   = eattr + (size_t)t * EE;
    const float* xt  = x + (size_t)t * NN * DIN;

    // graph normalization
    fill_kernel<<<gN, B, 0, stream>>>(deg, NN, 0.f);
    edge_deg_kernel<<<gE, B, 0, stream>>>(src, w, deg);
    dinv_kernel<<<gN, B, 0, stream>>>(deg);
    wn_kernel<<<gE, B, 0, stream>>>(src, dst, w, deg, wn);

    // r/u path: comb = [x_t, h]
    build_comb_kernel<<<gNF, B, 0, stream>>>(xt, h, nullptr, 0, C);
    fill_kernel<<<gNF, B, 0, stream>>>(T1, NN * FPAD, 0.f);
    spmm_kernel<<<gSp, B, 0, stream>>>(src, dst, wn, C, T1);
    fill_kernel<<<gNF, B, 0, stream>>>(T2, NN * FPAD, 0.f);
    spmm_kernel<<<gSp, B, 0, stream>>>(src, dst, wn, T1, T2);
    finish_t2_kernel<<<gNF, B, 0, stream>>>(T2, C);
    gemm_ru_kernel<<<gGm, B, 0, stream>>>(C, T1, T2, Wr, br, Wu, bu, rG, uG);

    // c path: comb_c = [x_t, r*h] (reuses C buffer)
    build_comb_kernel<<<gNF, B, 0, stream>>>(xt, h, rG, 1, C);
    fill_kernel<<<gNF, B, 0, stream>>>(T1, NN * FPAD, 0.f);
    spmm_kernel<<<gSp, B, 0, stream>>>(src, dst, wn, C, T1);
    fill_kernel<<<gNF, B, 0, stream>>>(T2, NN * FPAD, 0.f);
    spmm_kernel<<<gSp, B, 0, stream>>>(src, dst, wn, T1, T2);
    finish_t2_kernel<<<gNF, B, 0, stream>>>(T2, C);
    gemm_c_kernel<<<gGm, B, 0, stream>>>(C, T1, T2, Wc, bc, uG, h);
  }
}